// DynamicTinyOBBAssigner_80960133529823
// MI455X (gfx1250) — compile-verified
//
#include <hip/hip_runtime.h>
#include <math.h>

// Problem constants (match reference)
#define BB 16
#define NN 8400
#define GG 128
#define NC 5
#define TOPK_ 13
#define MAXT 18            // TOPK + TINY_BOOST + 1
#define MAXFB 8            // 3 + TINY_BOOST + 1
#define GTW 20             // words per gt descriptor
#define EPSF 1e-9f
#define PI_F 3.14159265358979323846f
#define HALF_PI_F 1.57079632679489661923f
#define INV_PI_F 0.31830988618379067154f
#define RADIUS_F 1.5f

typedef float v2f __attribute__((ext_vector_type(2)));
typedef float v8f __attribute__((ext_vector_type(8)));

__device__ __forceinline__ int imin_(int a, int b) { return a < b ? a : b; }
__device__ __forceinline__ int imax_(int a, int b) { return a > b ? a : b; }

// ---------------- zero scratch ----------------
__global__ void k_zero(unsigned* __restrict__ p, int n) {
  int i = blockIdx.x * 256 + threadIdx.x;
  if (i < n) p[i] = 0u;
}

// ---------------- precompute ALPHA*log(clip(sigmoid)+EPS) ----------------
__global__ void k_cls(const float* __restrict__ ps, float* __restrict__ clsA) {
  int i = blockIdx.x * 256 + threadIdx.x;
  if (i >= BB * NN * NC) return;
  float s = 1.0f / (1.0f + expf(-ps[i]));
  s = fminf(fmaxf(s, EPSF), 1.0f);
  clsA[i] = 0.5f * logf(s + EPSF);
}

// ---------------- per-gt derived constants ----------------
__global__ void k_gt(const float* __restrict__ gtb, const int* __restrict__ gtl,
                     const float* __restrict__ mg, float* __restrict__ gtw) {
  int i = blockIdx.x * 256 + threadIdx.x;
  if (i >= BB * GG) return;
  const float* g5 = gtb + i * 5;
  float cx = g5[0], cy = g5[1], w = g5[2], h = g5[3], th = g5[4];
  float wc = fmaxf(w, EPSF), hc = fmaxf(h, EPSF);
  float area = wc * hc;
  float ar = fmaxf(wc / hc, hc / wc);
  float tiny = fminf(fmaxf((0.01f - area) / (0.01f + EPSF), 0.0f), 1.0f);
  int extra = (int)ceilf(tiny * 4.0f);
  int elong = (ar >= 4.0f) ? 1 : 0;
  int budget = imin_(imax_(TOPK_ + extra + elong, 1), NN);
  float scale = sqrtf(w * h);            // raw wh, per reference _box_metrics
  float inv_s = 1.0f / (scale + 1e-6f);
  float inv_s15 = 1.0f / (scale * 1.5f + 1e-6f);
  float tsc = 1.0f + tiny * (4.0f / 13.0f);
  float str = fminf(fmaxf((ar - 4.0f) / (4.0f + EPSF), 0.0f), 1.0f);
  float mult = tsc * (1.0f + 0.25f * str);
  float ct = cosf(th), st = -sinf(th);   // cos(-th), sin(-th)
  float* o = gtw + i * GTW;
  o[0] = cx; o[1] = cy; o[2] = w; o[3] = h; o[4] = th;
  o[5] = ct; o[6] = st; o[7] = w * RADIUS_F; o[8] = h * RADIUS_F;
  o[9] = inv_s; o[10] = inv_s15; o[11] = mult;
  o[12] = mg[i];
  o[13] = __int_as_float(gtl[i]);
  o[14] = __int_as_float(budget);
  o[15] = __int_as_float(imax_(budget - TOPK_ + 3, 3));       // fallback_k
  o[16] = __int_as_float(((area < 0.01f) || (ar >= 4.0f)) ? 1 : 0);
  o[17] = 0.0f; o[18] = 0.0f; o[19] = 0.0f;
}

// ---------------- tiny fallback: closest-k anchors for protected gts ----------------
__global__ __launch_bounds__(256) void k_fb(const float* __restrict__ ap,
                                            const float* __restrict__ gtw,
                                            unsigned* __restrict__ fbbits) {
  int bg = blockIdx.x;
  int b = bg / GG, g = bg % GG;
  const float* G_ = gtw + bg * GTW;
  float mask = G_[12];
  int prot_cand = __float_as_int(G_[16]);
  if (mask <= 0.0f || prot_cand == 0) return;      // uniform per block
  float cx = G_[0], cy = G_[1], ct = G_[5], st = G_[6], wR = G_[7], hR = G_[8];
  int fbk = __float_as_int(G_[15]);

  __shared__ float sd[NN];
  __shared__ int scnt[256];
  __shared__ float rv[256];
  __shared__ int ri[256];

  int tid = threadIdx.x;
  int cnt = 0;
  for (int n = tid; n < NN; n += 256) {
    float ax = ap[2 * n], ay = ap[2 * n + 1];
    float dx = ax - cx, dy = ay - cy;
    float lx = dx * ct - dy * st;
    float ly = dx * st + dy * ct;
    if ((fabsf(lx) < wR) && (fabsf(ly) < hR)) cnt++;
    sd[n] = dx * dx + dy * dy;
  }
  scnt[tid] = cnt;
  __syncthreads();
  for (int s = 128; s > 0; s >>= 1) {
    if (tid < s) scnt[tid] += scnt[tid + s];
    __syncthreads();
  }
  if (scnt[0] != 0) return;   // uniform: count over n nonzero -> not protected

  for (int k = 0; k < fbk; k++) {
    float bv = 3.3e38f; int bi = NN;
    for (int n = tid; n < NN; n += 256) {
      float v = sd[n];
      if (v < bv || (v == bv && n < bi)) { bv = v; bi = n; }
    }
    rv[tid] = bv; ri[tid] = bi;
    __syncthreads();
    for (int s = 128; s > 0; s >>= 1) {
      if (tid < s) {
        if (rv[tid + s] < rv[tid] || (rv[tid + s] == rv[tid] && ri[tid + s] < ri[tid])) {
          rv[tid] = rv[tid + s]; ri[tid] = ri[tid + s];
        }
      }
      __syncthreads();
    }
    if (tid == 0) {
      int win = ri[0];
      sd[win] = 3.4e38f;
      atomicOr(&fbbits[((size_t)b * NN + win) * 4 + (g >> 5)], 1u << (g & 31));
    }
    __syncthreads();
  }
}

// ---------------- per-(b,g) top-18 + positive scatter ----------------
__global__ __launch_bounds__(256) void k_top(const float* __restrict__ pb,
                                             const float* __restrict__ ap,
                                             const float* __restrict__ clsA,
                                             const float* __restrict__ gtw,
                                             const unsigned* __restrict__ fbbits,
                                             unsigned* __restrict__ posflag,
                                             unsigned* __restrict__ normbuf) {
  int bg = blockIdx.x;
  int b = bg / GG, g = bg % GG;
  const float* G_ = gtw + bg * GTW;
  float mask = G_[12];
  if (mask <= 0.0f) return;   // row contributes no positives
  float cx = G_[0], cy = G_[1], th = G_[4], ct = G_[5], st = G_[6];
  float wR = G_[7], hR = G_[8], inv_s = G_[9], inv_s15 = G_[10], mult = G_[11];
  int label = __float_as_int(G_[13]);
  int budget = __float_as_int(G_[14]);

  __shared__ float sam[NN];
  __shared__ float rv[256];
  __shared__ int ri[256];
  __shared__ float topv[MAXT];
  __shared__ int topi[MAXT];

  int tid = threadIdx.x;
  int wword = g >> 5;
  unsigned bitm = 1u << (g & 31);

  for (int n = tid; n < NN; n += 256) {
    float ax = ap[2 * n], ay = ap[2 * n + 1];
    float dxa = ax - cx, dya = ay - cy;
    float lx = dxa * ct - dya * st;
    float ly = dxa * st + dya * ct;
    bool inc = (fabsf(lx) < wR) && (fabsf(ly) < hR);
    if (!inc) inc = (fbbits[((size_t)b * NN + n) * 4 + wword] & bitm) != 0u;
    float am = 0.0f;
    if (inc) {
      const float* pbn = pb + ((size_t)b * NN + n) * 5;
      float dx = pbn[0] - cx, dy = pbn[1] - cy;
      float ov = expf(-(dx * dx + dy * dy) * inv_s * inv_s);
      float lo = logf(ov + EPSF);
      float A = clsA[((size_t)b * NN + n) * NC + label];
      float x = pbn[4] - th + HALF_PI_F;
      float mm = x - floorf(x * INV_PI_F) * PI_F;           // jnp.remainder(x, pi)
      float ang = expf(-1.5f * fabsf(mm - HALF_PI_F));
      float cen = expf(-(dxa * dxa + dya * dya) * inv_s15 * inv_s15);
      am = expf(A + 6.0f * lo) * ang * mult * cen;          // mask == 1 here
    }
    sam[n] = am;
  }
  __syncthreads();

  for (int k = 0; k < MAXT; k++) {
    float bv = -1.0f; int bi = NN;
    for (int n = tid; n < NN; n += 256) {
      float v = sam[n];
      if (v > bv || (v == bv && n < bi)) { bv = v; bi = n; }
    }
    rv[tid] = bv; ri[tid] = bi;
    __syncthreads();
    for (int s = 128; s > 0; s >>= 1) {
      if (tid < s) {
        if (rv[tid + s] > rv[tid] || (rv[tid + s] == rv[tid] && ri[tid + s] < ri[tid])) {
          rv[tid] = rv[tid + s]; ri[tid] = ri[tid + s];
        }
      }
      __syncthreads();
    }
    if (tid == 0) {
      topv[k] = fmaxf(rv[0], 0.0f);
      topi[k] = ri[0];
      sam[ri[0]] = -2.0f;   // exclude from later rounds
    }
    __syncthreads();
  }

  if (tid == 0) {
    float tov[MAXT];
    float sum = 0.0f;
    for (int k = 0; k < MAXT; k++) {
      int n = topi[k];
      const float* pbn = pb + ((size_t)b * NN + n) * 5;
      float dx = pbn[0] - cx, dy = pbn[1] - cy;
      float ov = expf(-(dx * dx + dy * dy) * inv_s * inv_s);
      ov = fminf(fmaxf(ov, 0.0f), 1.0f);
      tov[k] = ov;
      sum += ov;
    }
    int dynk = (int)rintf(sum);     // RNE matches jnp.round
    dynk = imin_(imax_(dynk, 1), budget);
    float pam = 0.0f, pom = 0.0f;
    bool any = false;
    for (int k = 0; k < MAXT; k++) {
      if ((topv[k] > EPSF) && (k < dynk)) {
        pam = fmaxf(pam, topv[k]);
        pom = fmaxf(pom, tov[k]);
        any = true;
      }
    }
    if (any) {
      float ratio = pom / (pam + EPSF);
      for (int k = 0; k < MAXT; k++) {
        if ((topv[k] > EPSF) && (k < dynk)) {
          int n = topi[k];
          float v = topv[k] * ratio;   // v >= 0 -> uint atomicMax is order-preserving
          atomicMax(&normbuf[(size_t)b * NN + n], __float_as_uint(v));
          atomicOr(&posflag[(size_t)b * NN + n], 1u);
        }
      }
    }
  }
}

// ---------------- outputs: WMMA 16x16x4 f32 pairwise d^2 + argmax over g ----------------
__global__ __launch_bounds__(256) void k_out(const float* __restrict__ pb,
                                             const float* __restrict__ ap,
                                             const float* __restrict__ clsA,
                                             const float* __restrict__ gtw,
                                             const unsigned* __restrict__ fbbits,
                                             const unsigned* __restrict__ posflag,
                                             const float* __restrict__ normbuf,
                                             float* __restrict__ out) {
  const int b = blockIdx.y;
  __shared__ float sg[GG * GTW];
  for (int i = threadIdx.x; i < GG * GTW; i += 256) sg[i] = gtw[(size_t)b * GG * GTW + i];
  __syncthreads();

  const int wave = threadIdx.x >> 5;
  const int lane = threadIdx.x & 31;
  const int tileBase = blockIdx.x * 128 + wave * 16;
  if (tileBase >= NN) return;   // whole wave exits together (EXEC all-ones for WMMA)

  const int col = lane & 15;
  const bool hi = lane >= 16;

  // A operands: lane<16 -> row col holds [x, y]; lane>=16 -> row col holds [x*x+y*y, 1]
  int aA = imin_(tileBase + col, NN - 1);
  float pbx = pb[((size_t)b * NN + aA) * 5 + 0];
  float pby = pb[((size_t)b * NN + aA) * 5 + 1];
  v2f Ap; Ap.x = hi ? (pbx * pbx + pby * pby) : pbx; Ap.y = hi ? 1.0f : pby;
  float axl = ap[2 * aA], ayl = ap[2 * aA + 1];
  v2f Aa; Aa.x = hi ? (axl * axl + ayl * ayl) : axl; Aa.y = hi ? 1.0f : ayl;

  // per-row (anchor) registers: this lane owns rows rowsBase..rowsBase+7 of the C tile
  const int rowsBase = tileBase + (hi ? 8 : 0);
  float axr[8], ayr[8], pthr[8];
#pragma unroll
  for (int r = 0; r < 8; r++) {
    int n = imin_(rowsBase + r, NN - 1);
    axr[r] = ap[2 * n];
    ayr[r] = ap[2 * n + 1];
    pthr[r] = pb[((size_t)b * NN + n) * 5 + 4];
  }

  float rmax[8]; int ridx[8];
#pragma unroll
  for (int r = 0; r < 8; r++) { rmax[r] = -1.0f; ridx[r] = 0; }

  for (int gb = 0; gb < GG / 16; gb++) {
    const int g = gb * 16 + col;
    const float* Gp = &sg[g * GTW];
    float cx = Gp[0], cy = Gp[1];
    // B operand: lane<16 -> K0,K1 rows = [-2cx, -2cy]; lane>=16 -> K2,K3 rows = [1, cx^2+cy^2]
    v2f Bg; Bg.x = hi ? 1.0f : (-2.0f * cx); Bg.y = hi ? (cx * cx + cy * cy) : (-2.0f * cy);
    v8f c0 = {};
    v8f d2p = __builtin_amdgcn_wmma_f32_16x16x4_f32(false, Ap, false, Bg, (short)0, c0, false, false);
    v8f d2a = __builtin_amdgcn_wmma_f32_16x16x4_f32(false, Aa, false, Bg, (short)0, c0, false, false);

    float th = Gp[4], ct = Gp[5], st = Gp[6], wR = Gp[7], hR = Gp[8];
    float inv_s = Gp[9], inv_s15 = Gp[10], mult = Gp[11], mask = Gp[12];
    int label = __float_as_int(Gp[13]);
    unsigned bitm = 1u << (g & 31);
    int wword = g >> 5;

#pragma unroll
    for (int r = 0; r < 8; r++) {
      int n = rowsBase + r;
      float am = 0.0f;
      if (mask > 0.0f && n < NN) {
        float dxa = axr[r] - cx, dya = ayr[r] - cy;
        float lx = dxa * ct - dya * st;
        float ly = dxa * st + dya * ct;
        bool inc = (fabsf(lx) < wR) && (fabsf(ly) < hR);
        if (!inc) inc = (fbbits[((size_t)b * NN + n) * 4 + wword] & bitm) != 0u;
        if (inc) {
          float ov = expf(-fmaxf(d2p[r], 0.0f) * inv_s * inv_s);
          float lo = logf(ov + EPSF);
          float A = clsA[((size_t)b * NN + n) * NC + label];
          float x = pthr[r] - th + HALF_PI_F;
          float mm = x - floorf(x * INV_PI_F) * PI_F;
          float ang = expf(-1.5f * fabsf(mm - HALF_PI_F));
          float cen = expf(-fmaxf(d2a[r], 0.0f) * inv_s15 * inv_s15);
          am = expf(A + 6.0f * lo) * ang * mult * cen;
        }
      }
      if (am > rmax[r] || (am == rmax[r] && g < ridx[r])) { rmax[r] = am; ridx[r] = g; }
    }
  }

  // reduce (max, first-index) across the 16 lanes holding the g-columns of each row
#pragma unroll
  for (int off = 8; off >= 1; off >>= 1) {
#pragma unroll
    for (int r = 0; r < 8; r++) {
      float ov_ = __shfl_xor(rmax[r], off, 16);
      int oi_ = __shfl_xor(ridx[r], off, 16);
      if (ov_ > rmax[r] || (ov_ == rmax[r] && oi_ < ridx[r])) { rmax[r] = ov_; ridx[r] = oi_; }
    }
  }

  if (col == 0) {
    const size_t L0 = 0;
    const size_t L1 = (size_t)BB * NN;
    const size_t L2 = (size_t)BB * NN * 6;
    const size_t L3 = (size_t)BB * NN * 11;
#pragma unroll
    for (int r = 0; r < 8; r++) {
      int n = rowsBase + r;
      if (n >= NN) continue;
      size_t idx = (size_t)b * NN + n;
      unsigned pf = posflag[idx];
      float nrm = normbuf[idx];
      int a2g = ridx[r];
      const float* Gp = &sg[a2g * GTW];
      int lbl = pf ? __float_as_int(Gp[13]) : NC;
      out[L0 + idx] = (float)lbl;
      float* ob = out + L1 + idx * 5;
      ob[0] = Gp[0]; ob[1] = Gp[1]; ob[2] = Gp[2]; ob[3] = Gp[3]; ob[4] = Gp[4];
      float* os = out + L2 + idx * 5;
#pragma unroll
      for (int c = 0; c < NC; c++) os[c] = (pf && c == lbl) ? nrm : 0.0f;
      out[L3 + idx] = pf ? 1.0f : 0.0f;
    }
  }
}

// ---------------- launcher ----------------
extern "C" void kernel_launch(void* const* d_in, const int* in_sizes, int n_in,
                              void* d_out, int out_size, void* d_ws, size_t ws_size,
                              hipStream_t stream) {
  (void)in_sizes; (void)n_in; (void)out_size; (void)ws_size;
  const float* pred_scores = (const float*)d_in[0];
  const float* pred_bboxes = (const float*)d_in[1];
  const float* anchor_points = (const float*)d_in[2];
  const int* gt_labels = (const int*)d_in[3];
  const float* gt_bboxes = (const float*)d_in[4];
  const float* mask_gt = (const float*)d_in[5];
  float* out = (float*)d_out;

  char* ws = (char*)d_ws;
  size_t off = 0;
  float* clsA = (float*)(ws + off);        off += (size_t)BB * NN * NC * 4;  off = (off + 255) & ~(size_t)255;
  float* gtw = (float*)(ws + off);         off += (size_t)BB * GG * GTW * 4; off = (off + 255) & ~(size_t)255;
  // contiguous zero-init region: fbbits | normbuf | posflag
  unsigned* fbbits = (unsigned*)(ws + off);   off += (size_t)BB * NN * 4 * 4;
  unsigned* normbuf = (unsigned*)(ws + off);  off += (size_t)BB * NN * 4;
  unsigned* posflag = (unsigned*)(ws + off);  off += (size_t)BB * NN * 4;

  const int nz = BB * NN * 6;
  k_zero<<<(nz + 255) / 256, 256, 0, stream>>>(fbbits, nz);
  k_cls<<<(BB * NN * NC + 255) / 256, 256, 0, stream>>>(pred_scores, clsA);
  k_gt<<<(BB * GG + 255) / 256, 256, 0, stream>>>(gt_bboxes, gt_labels, mask_gt, gtw);
  k_fb<<<BB * GG, 256, 0, stream>>>(anchor_points, gtw, fbbits);
  k_top<<<BB * GG, 256, 0, stream>>>(pred_bboxes, anchor_points, clsA, gtw, fbbits, posflag, normbuf);
  dim3 gOut((NN + 127) / 128, BB);
  k_out<<<gOut, 256, 0, stream>>>(pred_bboxes, anchor_points, clsA, gtw, fbbits, posflag,
                                  (const float*)normbuf, out);
}